// KBestDetector_64828236366364
// MI455X (gfx1250) — compile-verified
//
#include <hip/hip_runtime.h>
#include <hip/hip_bf16.h>

typedef __attribute__((ext_vector_type(2))) float v2f;
typedef __attribute__((ext_vector_type(8))) float v8f;

#define WAVES_PER_BLOCK 8
#define BIGD 1.0e9f
#define HUGED 3.0e38f

// Per-wave scratch. "big" is reused: front-end holds Sr/Si (16x16) and Hr/Hi
// (16x8); K-best phase holds the 1024-entry candidate-distance matrix.
struct WaveScratch {
  float big[1024];
  float Rr[64], Ri[64];
  float ytr[16], yti[16];
  float ybr[8], ybi[8];
  float normv[8];
  int   order[8];
  float dists[2][64];
  unsigned syms[2][64];
  float crd[64], cid[64], td[64];
  float pr[16], pi[16];
};

__device__ __forceinline__ float waveSum(float v) {
#pragma unroll
  for (int off = 16; off; off >>= 1) v += __shfl_xor(v, off, 32);
  return v;
}

__global__ void __launch_bounds__(32 * WAVES_PER_BLOCK)
kbest_kernel(const float* __restrict__ yr, const float* __restrict__ yi,
             const float* __restrict__ hr, const float* __restrict__ hi,
             const float* __restrict__ sr, const float* __restrict__ si,
             const float* __restrict__ ptsr, const float* __restrict__ ptsi,
             float* __restrict__ out, int B) {
  __shared__ WaveScratch wsd[WAVES_PER_BLOCK];
  const int wave = threadIdx.x >> 5;
  const int lane = threadIdx.x & 31;
  const int b = blockIdx.x * WAVES_PER_BLOCK + wave;
  if (b >= B) return;
  WaveScratch& w = wsd[wave];
  float* Sr = w.big;        // 16x16
  float* Si = w.big + 256;  // 16x16
  float* Hr = w.big + 512;  // 16x8
  float* Hi = w.big + 640;  // 16x8

  // ---------------- load inputs (wave-private, no cross-wave sync needed) ---
  for (int t = lane; t < 256; t += 32) { Sr[t] = sr[b * 256 + t]; Si[t] = si[b * 256 + t]; }
  for (int t = lane; t < 128; t += 32) { Hr[t] = hr[b * 128 + t]; Hi[t] = hi[b * 128 + t]; }
  if (lane < 16) {
    w.ytr[lane] = yr[b * 16 + lane];
    w.yti[lane] = yi[b * 16 + lane];
    w.pr[lane] = ptsr[lane];
    w.pi[lane] = ptsi[lane];
  }

  // ---------------- complex Cholesky, in place (diag real positive) --------
  for (int j = 0; j < 16; ++j) {
    float acc = 0.f;
    for (int k = 0; k < j; ++k) { float a = Sr[j * 16 + k], c = Si[j * 16 + k]; acc += a * a + c * c; }
    float diag = sqrtf(fmaxf(Sr[j * 16 + j] - acc, 1e-20f));
    if (lane == j) { Sr[j * 16 + j] = diag; Si[j * 16 + j] = 0.f; }
    const int i = lane;
    if (i > j && i < 16) {
      float sre = 0.f, sim = 0.f;
      for (int k = 0; k < j; ++k) {
        float ar = Sr[i * 16 + k], ai = Si[i * 16 + k];
        float br = Sr[j * 16 + k], bi = Si[j * 16 + k];
        sre += ar * br + ai * bi;   // a * conj(b)
        sim += ai * br - ar * bi;
      }
      float inv = 1.f / diag;
      Sr[i * 16 + j] = (Sr[i * 16 + j] - sre) * inv;
      Si[i * 16 + j] = (Si[i * 16 + j] - sim) * inv;
    }
  }

  // ---------------- forward substitution: y_t = L^-1 y ---------------------
  for (int i = 0; i < 16; ++i) {
    float inv = 1.f / Sr[i * 16 + i];
    float vr = w.ytr[i] * inv, vi = w.yti[i] * inv;
    if (lane == i) { w.ytr[i] = vr; w.yti[i] = vi; }
    const int r = lane;
    if (r > i && r < 16) {
      float lr = Sr[r * 16 + i], li = Si[r * 16 + i];
      w.ytr[r] -= lr * vr - li * vi;
      w.yti[r] -= lr * vi + li * vr;
    }
  }

  // ---------------- H_t = L^-1 H, two columns per pass (half-wave each) ----
  for (int cc = 0; cc < 4; ++cc) {
    const int col = cc * 2 + (lane >> 4);
    const int r = lane & 15;
    for (int i = 0; i < 16; ++i) {
      float inv = 1.f / Sr[i * 16 + i];
      float vr = Hr[i * 8 + col] * inv, vi = Hi[i * 8 + col] * inv;
      if (r == i) { Hr[i * 8 + col] = vr; Hi[i * 8 + col] = vi; }
      if (r > i) {
        float lr = Sr[r * 16 + i], li = Si[r * 16 + i];
        Hr[r * 8 + col] -= lr * vr - li * vi;
        Hi[r * 8 + col] -= lr * vi + li * vr;
      }
    }
  }

  // ---------------- column norms, descending order --------------------------
  if (lane < 8) {
    float s2 = 0.f;
    for (int m = 0; m < 16; ++m) { float a = Hr[m * 8 + lane], c = Hi[m * 8 + lane]; s2 += a * a + c * c; }
    w.normv[lane] = s2;
  }
  if (lane == 0) {
    unsigned used = 0;
    for (int j = 0; j < 8; ++j) {
      int best = 0; float bv = -1.f;
      for (int s2 = 0; s2 < 8; ++s2)
        if (!((used >> s2) & 1) && w.normv[s2] > bv) { bv = w.normv[s2]; best = s2; }
      used |= 1u << best;
      w.order[j] = best;
    }
  }

  // ---------------- MGS QR on permuted columns; build R, y_bar --------------
  for (int j = 0; j < 8; ++j) {
    const int cj = w.order[j];
    float v = 0.f;
    if (lane < 16) { float a = Hr[lane * 8 + cj], c = Hi[lane * 8 + cj]; v = a * a + c * c; }
    v = waveSum(v);
    float rjj = sqrtf(fmaxf(v, 1e-20f));
    if (lane == 0) { w.Rr[j * 8 + j] = rjj; w.Ri[j * 8 + j] = 0.f; }
    float inv = 1.f / rjj;
    if (lane < 16) { Hr[lane * 8 + cj] *= inv; Hi[lane * 8 + cj] *= inv; }
    float sre = 0.f, sim = 0.f;
    if (lane < 16) {
      float qr_ = Hr[lane * 8 + cj], qi_ = Hi[lane * 8 + cj];
      float yr_ = w.ytr[lane], yi_ = w.yti[lane];
      sre = qr_ * yr_ + qi_ * yi_;   // conj(q)*y
      sim = qr_ * yi_ - qi_ * yr_;
    }
    sre = waveSum(sre); sim = waveSum(sim);
    if (lane == 0) { w.ybr[j] = sre; w.ybi[j] = sim; }
    for (int k = j + 1; k < 8; ++k) {
      const int ck = w.order[k];
      float ar = 0.f, ai = 0.f;
      if (lane < 16) {
        float qr_ = Hr[lane * 8 + cj], qi_ = Hi[lane * 8 + cj];
        float hre = Hr[lane * 8 + ck], him = Hi[lane * 8 + ck];
        ar = qr_ * hre + qi_ * him;
        ai = qr_ * him - qi_ * hre;
      }
      ar = waveSum(ar); ai = waveSum(ai);
      if (lane == 0) { w.Rr[j * 8 + k] = ar; w.Ri[j * 8 + k] = ai; }
      if (lane < 16) {
        float qr_ = Hr[lane * 8 + cj], qi_ = Hi[lane * 8 + cj];
        Hr[lane * 8 + ck] -= ar * qr_ - ai * qi_;
        Hi[lane * 8 + ck] -= ar * qi_ + ai * qr_;
      }
    }
  }

  // ---------------- K-best search ------------------------------------------
  for (int kk = 0; kk < 2; ++kk) {
    int k = lane + kk * 32;
    w.dists[0][k] = (k == 0) ? 0.f : BIGD;
    w.syms[0][k] = 0u;
  }
  int buf = 0;
  const int q = lane & 15;
  const float prq = w.pr[q], piq = w.pi[q];

  for (int l = 7; l >= 0; --l) {
    const float Rll = w.Rr[l * 8 + l];
    const float aqr = Rll * prq, aqi = Rll * piq;
    const float wq = aqr * aqr + aqi * aqi;
    // B(4x16) rows = {a_r, a_i, 1, w}; K splits at lane-16 midpoint.
    v2f Bv;
    Bv.x = (lane < 16) ? aqr : 1.0f;
    Bv.y = (lane < 16) ? aqi : wq;

    const float ybr_l = w.ybr[l], ybi_l = w.ybi[l];
    float* dcur = w.dists[buf];
    unsigned* scur = w.syms[buf];
    // c_k = y_bar[l] - sum_{j>l} R[l][j]*p(sym_k[j]); two candidates per lane
    for (int kk = 0; kk < 2; ++kk) {
      int k = lane + kk * 32;
      float cr = ybr_l, ci = ybi_l;
      unsigned sy = scur[k];
      for (int j = l + 1; j < 8; ++j) {
        int s4 = (int)((sy >> (4 * j)) & 15u);
        float rr = w.Rr[l * 8 + j], ri = w.Ri[l * 8 + j];
        float xr = w.pr[s4], xi = w.pi[s4];
        cr -= rr * xr - ri * xi;
        ci -= rr * xi + ri * xr;
      }
      w.crd[k] = cr; w.cid[k] = ci;
      w.td[k] = dcur[k] + cr * cr + ci * ci;
    }

    // d_new[k][q] = td[k] + w[q] - 2(cr*ar + ci*ai)  as  (16x4)x(4x16) WMMA.
    const int m = lane & 15;
    for (int g = 0; g < 4; ++g) {
      float c0 = w.crd[g * 16 + m], c1 = w.cid[g * 16 + m], t0 = w.td[g * 16 + m];
      v2f Av;                              // A(16x4) rows = {-2cr, -2ci, t, 1}
      Av.x = (lane < 16) ? (-2.f * c0) : t0;
      Av.y = (lane < 16) ? (-2.f * c1) : 1.0f;
      v8f acc = {0.f, 0.f, 0.f, 0.f, 0.f, 0.f, 0.f, 0.f};
#if __has_builtin(__builtin_amdgcn_wmma_f32_16x16x4_f32)
      acc = __builtin_amdgcn_wmma_f32_16x16x4_f32(false, Av, false, Bv,
                                                  (short)0, acc, false, false);
#pragma unroll
      for (int vv = 0; vv < 8; ++vv) {
        int cand = g * 16 + vv + ((lane < 16) ? 0 : 8);
        w.big[cand * 16 + q] = acc[vv];
      }
#else
#pragma unroll
      for (int vv = 0; vv < 8; ++vv) {
        int cand = g * 16 + vv + ((lane < 16) ? 0 : 8);
        w.big[cand * 16 + q] =
            w.td[cand] + wq - 2.f * (w.crd[cand] * aqr + w.cid[cand] * aqi);
      }
#endif
    }

    // exact top-64 (smallest) by repeated min-extraction; ties -> lower index
    float* dnext = w.dists[buf ^ 1];
    unsigned* snext = w.syms[buf ^ 1];
    for (int t = 0; t < 64; ++t) {
      float mv = HUGED; int mi = 1 << 30;
      for (int i2 = 0; i2 < 32; ++i2) {
        int idx = i2 * 32 + lane;
        float v2 = w.big[idx];
        if (v2 < mv) { mv = v2; mi = idx; }
      }
#pragma unroll
      for (int off = 16; off; off >>= 1) {
        float ov = __shfl_xor(mv, off, 32);
        int oi = __shfl_xor(mi, off, 32);
        if (ov < mv || (ov == mv && oi < mi)) { mv = ov; mi = oi; }
      }
      if (lane == (mi & 31)) w.big[mi] = HUGED;
      if (lane == 0) {
        int par = mi >> 4, qq = mi & 15;
        dnext[t] = mv;
        snext[t] = (scur[par] & ~(0xFu << (4 * l))) | ((unsigned)qq << (4 * l));
      }
    }
    buf ^= 1;
  }

  // ---------------- LLRs: 32 (stream,bit) pairs == 32 lanes -----------------
  {
    float* dfin = w.dists[buf];
    unsigned* sfin = w.syms[buf];
    const int j = lane >> 2;    // sorted stream position
    const int bpos = lane & 3;  // bit (MSB first)
    float m0 = BIGD, m1 = BIGD;
    for (int k = 0; k < 64; ++k) {
      unsigned nib = (sfin[k] >> (4 * j)) & 15u;
      int bit = (int)((nib >> (3 - bpos)) & 1u);
      float d = dfin[k];
      if (bit == 0) m0 = fminf(m0, d); else m1 = fminf(m1, d);
    }
    float llr = fminf(fmaxf(m0 - m1, -20.f), 20.f);
    int orow = w.order[j];  // inverse permutation scatter
    out[b * 32 + orow * 4 + bpos] = llr;
  }
}

extern "C" void kernel_launch(void* const* d_in, const int* in_sizes, int n_in,
                              void* d_out, int out_size, void* d_ws, size_t ws_size,
                              hipStream_t stream) {
  const float* yr = (const float*)d_in[0];
  const float* yi = (const float*)d_in[1];
  const float* hr = (const float*)d_in[2];
  const float* hi = (const float*)d_in[3];
  const float* sr = (const float*)d_in[4];
  const float* si = (const float*)d_in[5];
  const float* pr = (const float*)d_in[6];
  const float* pi = (const float*)d_in[7];
  int B = in_sizes[0] / 16;  // yr is (B, M=16)
  int blocks = (B + WAVES_PER_BLOCK - 1) / WAVES_PER_BLOCK;
  kbest_kernel<<<blocks, 32 * WAVES_PER_BLOCK, 0, stream>>>(
      yr, yi, hr, hi, sr, si, pr, pi, (float*)d_out, B);
}